// CropRoi_16527034155026
// MI455X (gfx1250) — compile-verified
//
#include <hip/hip_runtime.h>

// FPN ROI-Align (crop_and_resize, 14x14, 256 ch, 1024 rois) for gfx1250.
//
// Separable bilinear: per (proposal, y-row), y-blend two source rows in VALU,
// then x-interpolate as a 16x16x28 GEMM (M=16 channels, N=14 x's padded to 16,
// K=28 = 14 x-samples x 2 corners) via 7 chained V_WMMA_F32_16X16X4_F32.
//
// Round-2 tuning (from disasm): GVS addressing (uniform SGPR row base via
// readfirstlane + per-lane byte-offset VGPRs) kills per-load 64-bit VALU adds;
// paired b64 corner loads halve vmem ops; stores share one base address with
// constant IOFFSETs (r*784 bytes).

typedef float v2f __attribute__((ext_vector_type(2)));
typedef float v8f __attribute__((ext_vector_type(8)));

#define CS 14  // crop size

__global__ __launch_bounds__(128) void roialign_wmma_kernel(
    const float* __restrict__ p2, const float* __restrict__ p3,
    const float* __restrict__ p4, const float* __restrict__ p5,
    const float* __restrict__ props, float* __restrict__ out)
{
    const int prop = blockIdx.x;
    const int wave = threadIdx.x >> 5;        // 4 waves per block
    const int lane = threadIdx.x & 31;        // wave32
    const int ctile = blockIdx.y * 4 + wave;  // 16 tiles of 16 channels

    // ---- proposal params (uniform per block) ----
    const float* pr = props + prop * 7;
    const int   b  = (int)pr[0];
    const float x0 = pr[1], y0 = pr[2], x1 = pr[3], y1 = pr[4];

    // level = argmin |sqrt(w*h) - {8,16,32,64}| (first index wins ties)
    const float sz = sqrtf((x1 - x0) * (y1 - y0));
    int lvl = 0;
    float best = fabsf(sz - 8.0f);
    { float d = fabsf(sz - 16.0f); if (d < best) { best = d; lvl = 1; } }
    { float d = fabsf(sz - 32.0f); if (d < best) { best = d; lvl = 2; } }
    { float d = fabsf(sz - 64.0f); if (d < best) {           lvl = 3; } }
    lvl = __builtin_amdgcn_readfirstlane(lvl);   // force scalar level select

    const float scale = (lvl == 0) ? 0.25f : (lvl == 1) ? 0.125f
                      : (lvl == 2) ? 0.0625f : 0.03125f;
    const int   HW = 256 >> lvl;  // H == W per level: 256,128,64,32
    const float* f = (lvl == 0) ? p2 : (lvl == 1) ? p3 : (lvl == 2) ? p4 : p5;

    const float x0s = x0 * scale, x1s = x1 * scale;
    const float y0s = y0 * scale, y1s = y1 * scale;

    // ---- per-proposal x-sample tables (shared by all 4 waves) ----
    __shared__ float s_lx[CS];
    __shared__ int   s_xi0[CS];
    if (threadIdx.x < CS) {
        float g  = (float)threadIdx.x * (1.0f / 13.0f);
        float xs = fmaf(x1s - x0s, g, x0s);
        float xf = floorf(xs);
        s_lx[threadIdx.x]  = xs - xf;                       // frac of UNclamped floor
        s_xi0[threadIdx.x] = min(max((int)xf, 0), HW - 1);  // xi1 = min(xi0+1, HW-1)
    }
    __syncthreads();

    const int n  = lane & 15;   // A row (channel-in-tile), B/D col (x)
    const int hi = lane >> 4;   // K-half select

    // Per-lane, per-chunk precompute. Chunk k covers K={4k..4k+3}; this lane's
    // x-sample is t = 2k + hi.
    //   A lane pair = { yblend(col xi0[t]), yblend(col xi1[t]) }  (K even/odd)
    //   B lane pair = { (n==t)?(1-lx[t]):0 , (n==t)?lx[t]:0 }
    // Corner pair loaded as one b64 at col=min(xi0,HW-2); edge-clamped lanes
    // (xi0==HW-1) take the high element for both corners.
    const int ch = ctile * 16 + n;         // this lane's channel
    int  boff[7];                          // per-lane byte offset (y-invariant)
    bool edge[7];
    v2f  Bm[7];
#pragma unroll
    for (int k = 0; k < 7; ++k) {
        int t    = 2 * k + hi;
        int xi0  = s_xi0[t];
        int col  = min(xi0, HW - 2);
        boff[k]  = (ch * HW * HW + col) * 4;
        edge[k]  = (xi0 == HW - 1);
        float lx = s_lx[t];
        float sel = (n == t) ? 1.0f : 0.0f;
        Bm[k].x = sel * (1.0f - lx);
        Bm[k].y = sel * lx;
    }

    // Uniform (SGPR) base of this batch's feature map, in bytes
    const char* fb = (const char*)f + (size_t)b * 256u * HW * HW * 4u;

    // Store base: 8 channel rows at constant stride 196 floats (784 B IOFFSET)
    float* po = out + (size_t)(prop * 256 + ctile * 16 + hi * 8) * (CS * CS) + n;

    const float dy = y1s - y0s;
    for (int y = 0; y < CS; ++y) {
        // y sample (uniform across wave; readfirstlane -> SGPR row bases)
        float ys  = fmaf(dy, (float)y * (1.0f / 13.0f), y0s);
        float yf  = floorf(ys);
        float ly  = ys - yf;
        int  yi0  = __builtin_amdgcn_readfirstlane(min(max((int)yf, 0), HW - 1));
        int  yi1  = __builtin_amdgcn_readfirstlane(min(yi0 + 1, HW - 1));
        const char* row0 = fb + (size_t)(yi0 * HW) * 4;   // uniform SGPR64
        const char* row1 = fb + (size_t)(yi1 * HW) * 4;

        v8f acc = {0.f, 0.f, 0.f, 0.f, 0.f, 0.f, 0.f, 0.f};
#pragma unroll
        for (int k = 0; k < 7; ++k) {
            v2f q0, q1;                                   // saddr + voffset loads
            __builtin_memcpy(&q0, row0 + boff[k], 8);
            __builtin_memcpy(&q1, row1 + boff[k], 8);
            float v00 = edge[k] ? q0.y : q0.x;
            float v10 = edge[k] ? q1.y : q1.x;
            v2f a;
            a.x = fmaf(ly, v10 - v00, v00);               // (1-ly)*v00 + ly*v10
            a.y = fmaf(ly, q1.y - q0.y, q0.y);
            // D = A(16x4) x B(4x16) + C : accumulates x-interp over K=28
            acc = __builtin_amdgcn_wmma_f32_16x16x4_f32(
                false, a, false, Bm[k], (short)0, acc, false, false);
        }

        // D layout: VGPR r -> channel (r + 8*hi), lane&15 -> x
        float* pw = po + y * CS;
        if (n < CS) {
#pragma unroll
            for (int r = 0; r < 8; ++r)
                pw[r * (CS * CS)] = acc[r];               // const IOFFSET stores
        }
    }
}

extern "C" void kernel_launch(void* const* d_in, const int* in_sizes, int n_in,
                              void* d_out, int out_size, void* d_ws, size_t ws_size,
                              hipStream_t stream) {
    const float* p2    = (const float*)d_in[0];
    const float* p3    = (const float*)d_in[1];
    const float* p4    = (const float*)d_in[2];
    const float* p5    = (const float*)d_in[3];
    const float* props = (const float*)d_in[4];
    float* out = (float*)d_out;

    const int nprop = in_sizes[4] / 7;   // 1024
    // grid.y * 4 waves/block = 16 channel tiles of 16 (256 channels)
    dim3 grid(nprop, 4, 1);
    dim3 block(128, 1, 1);
    roialign_wmma_kernel<<<grid, block, 0, stream>>>(p2, p3, p4, p5, props, out);
}